// CurvLoss_41051297415804
// MI455X (gfx1250) — compile-verified
//
#include <hip/hip_runtime.h>
#include <hip/hip_bf16.h>
#include <math.h>

typedef __attribute__((ext_vector_type(2))) float v2f;
typedef __attribute__((ext_vector_type(8))) float v8f;

#define NPTS 4096   // points per cloud
#define NB   4      // batches
#define BQ   128    // query points per block (4 wave32)
#define TN   128    // ref points per chunk (8 col tiles)

// ---------------------------------------------------------------------------
// Fused kNN (+optional kappa) kernel.
//   K         : neighbors kept (16 for kappa passes, 1 for the 1-NN pass)
//   SKIP_SELF : exclude ref index == query index (kappa passes)
//   DO_KAPPA  : compute curvature from the K neighbors; else write argmin idx
//
// Selection key = |r|^2 - 2 q.r  (|q|^2 constant per query row, dropped).
// The key is produced ENTIRELY inside V_WMMA_F32_16X16X4_F32 by padding K=4:
//   A row = (xq, yq, zq, 1),  B col = (-2xr, -2yr, -2zr, |r|^2)
// so the accumulator holds the key directly (no epilogue FMA, no rr loads).
// B fragments are staged in LDS pre-swizzled into the exact 4x16 f32 WMMA
// layout: one ds_load_b64 per lane per tile, no selects, no exec churn.
// ---------------------------------------------------------------------------
template<int K, bool SKIP_SELF, bool DO_KAPPA>
__global__ __launch_bounds__(BQ)
void knn_kappa_kernel(const float* __restrict__ qpts,   // [B,3,N] query cloud
                      const float* __restrict__ rpts,   // [B,3,N] ref cloud
                      const float* __restrict__ nrm,    // [B,3,N] normals (kappa)
                      const int*   __restrict__ nidx,   // optional normal index map
                      float* __restrict__ out_kappa,    // [B,N]
                      int*   __restrict__ out_idx)      // [B,N]
{
    __shared__ v2f   s_bf[TN / 16][32];   // pre-swizzled B fragments (2 KB)
    __shared__ float s_key[TN][BQ];       // keys, [n][m]: contiguous-M b128 stores

    const int b      = blockIdx.y;
    const int qbase  = blockIdx.x * BQ;
    const int tid    = threadIdx.x;
    const int lane   = tid & 31;
    const int wave   = tid >> 5;
    const int lane15 = lane & 15;
    const bool hi    = (lane >= 16);

    const size_t base = (size_t)b * 3 * NPTS;
    const float* qx = qpts + base;
    const float* qy = qx + NPTS;
    const float* qz = qy + NPTS;
    const float* rx = rpts + base;
    const float* ry = rx + NPTS;
    const float* rz = ry + NPTS;

    // --- A fragments (load once): wave owns query rows [wave*32, wave*32+32)
    // 16x4 f32 A layout: lanes 0-15 hold (K0,K1)=(x,y); lanes 16-31 hold (K2,K3)=(z,1)
    v2f afrag[2];
#pragma unroll
    for (int t = 0; t < 2; ++t) {
        int gq = qbase + wave * 32 + t * 16 + lane15;
        float x = qx[gq], y = qy[gq], z = qz[gq];
        v2f a;
        a.x = hi ? z   : x;
        a.y = hi ? 1.f : y;
        afrag[t] = a;
    }

    // --- per-thread top-K state (register resident) ---
    float bd[K];
    int   bi[K];
#pragma unroll
    for (int i = 0; i < K; ++i) { bd[i] = 3.4e38f; bi[i] = 0; }
    float wmax = 3.4e38f;   // current worst kept key
    int   wslot = 0;

    const int myq = qbase + tid;   // this thread's query point

    for (int chunk = 0; chunk < NPTS; chunk += TN) {
        // stage ref chunk directly in B-fragment layout (every thread = 1 ref pt)
        {
            int gr = chunk + tid;
            float x = rx[gr], y = ry[gr], z = rz[gr];
            int c = tid >> 4, j = tid & 15;
            v2f lo; lo.x = -2.f * x; lo.y = -2.f * y;
            v2f hi2; hi2.x = -2.f * z; hi2.y = x * x + y * y + z * z;
            s_bf[c][j]      = lo;    // lanes 0-15: (K0,K1)
            s_bf[c][j + 16] = hi2;   // lanes 16-31: (K2,K3)
        }
        __syncthreads();

        // each wave: 2 row-tiles x 8 col-tiles; accumulator == selection key
#pragma unroll
        for (int t = 0; t < 2; ++t) {
#pragma unroll
            for (int c = 0; c < TN / 16; ++c) {
                v2f bf = s_bf[c][lane];          // single ds_load_b64
                v8f acc = {};
                acc = __builtin_amdgcn_wmma_f32_16x16x4_f32(
                    false, afrag[t], false, bf, (short)0, acc, false, false);
                // C/D layout: VGPR r -> M = r + (hi?8:0), N = lane15
                int nloc  = c * 16 + lane15;
                int mbase = wave * 32 + t * 16 + (hi ? 8 : 0);
#pragma unroll
                for (int r = 0; r < 8; ++r)
                    s_key[nloc][mbase + r] = acc[r];
            }
        }
        __syncthreads();

        // each thread scans its own query row of this chunk
#pragma unroll 8
        for (int n = 0; n < TN; ++n) {
            int gr = chunk + n;
            float key = s_key[n][tid];
            if (SKIP_SELF && gr == myq) continue;
            if (K == 1) {
                if (key < bd[0]) { bd[0] = key; bi[0] = gr; }
            } else {
                if (key < wmax) {
                    bd[wslot] = key; bi[wslot] = gr;
                    wmax = bd[0]; wslot = 0;
#pragma unroll
                    for (int i = 1; i < K; ++i)
                        if (bd[i] > wmax) { wmax = bd[i]; wslot = i; }
                }
            }
        }
        // no trailing barrier needed: next chunk's first __syncthreads orders
        // these s_key reads before the next WMMA-phase s_key writes.
    }

    if (DO_KAPPA) {
        float px = qx[myq], py = qy[myq], pz = qz[myq];
        int nn = nidx ? nidx[(size_t)b * NPTS + myq] : myq;
        const float* nb_ = nrm + base;
        float nxx = nb_[nn], nyy = nb_[NPTS + nn], nzz = nb_[2 * NPTS + nn];
        float acc = 0.f;
#pragma unroll
        for (int i = 0; i < K; ++i) {
            int g = bi[i];
            float dx = rx[g] - px, dy = ry[g] - py, dz = rz[g] - pz;
            float len = sqrtf(dx * dx + dy * dy + dz * dz);
            float inv = 1.0f / fmaxf(len, 1e-12f);
            acc += fabsf((dx * nxx + dy * nyy + dz * nzz) * inv);
        }
        out_kappa[(size_t)b * NPTS + myq] = acc * (1.0f / K);
    } else {
        out_idx[(size_t)b * NPTS + myq] = bi[0];
    }
}

// ---------------------------------------------------------------------------
// loss[b] = 10 * mean_n (adv_kappa[b,n] - kappa_ori[b, idx1[b,n]])^2
// ---------------------------------------------------------------------------
__global__ __launch_bounds__(256)
void curv_loss_kernel(const float* __restrict__ kori,
                      const float* __restrict__ kadv,
                      const int*   __restrict__ idx1,
                      float* __restrict__ out)
{
    __shared__ float red[256];
    int b = blockIdx.x;
    float s = 0.f;
    for (int n = threadIdx.x; n < NPTS; n += 256) {
        float d = kadv[b * NPTS + n] - kori[b * NPTS + idx1[b * NPTS + n]];
        s += d * d;
    }
    red[threadIdx.x] = s;
    __syncthreads();
    for (int off = 128; off > 0; off >>= 1) {
        if (threadIdx.x < off) red[threadIdx.x] += red[threadIdx.x + off];
        __syncthreads();
    }
    if (threadIdx.x == 0) out[b] = 10.0f * red[0] * (1.0f / NPTS);
}

extern "C" void kernel_launch(void* const* d_in, const int* in_sizes, int n_in,
                              void* d_out, int out_size, void* d_ws, size_t ws_size,
                              hipStream_t stream) {
    const float* pc_ori = (const float*)d_in[0];   // [B,3,N]
    const float* adv    = (const float*)d_in[1];   // [B,3,N]
    const float* nrm    = (const float*)d_in[2];   // [B,3,N]
    float* out = (float*)d_out;                    // [B]

    float* kori = (float*)d_ws;              // B*N floats
    float* kadv = kori + NB * NPTS;          // B*N floats
    int*   idx1 = (int*)(kadv + NB * NPTS);  // B*N ints

    dim3 grid(NPTS / BQ, NB);

    // 1-NN of adv into ori  -> idx1
    knn_kappa_kernel<1, false, false><<<grid, BQ, 0, stream>>>(
        adv, pc_ori, nullptr, nullptr, nullptr, idx1);
    // kappa of original cloud (own normals)
    knn_kappa_kernel<16, true, true><<<grid, BQ, 0, stream>>>(
        pc_ori, pc_ori, nrm, nullptr, kori, nullptr);
    // kappa of adversarial cloud (normals gathered via idx1)
    knn_kappa_kernel<16, true, true><<<grid, BQ, 0, stream>>>(
        adv, adv, nrm, idx1, kadv, nullptr);
    // final per-batch loss
    curv_loss_kernel<<<NB, 256, 0, stream>>>(kori, kadv, idx1, out);
}